// FGE_18047452577900
// MI455X (gfx1250) — compile-verified
//
#include <hip/hip_runtime.h>
#include <hip/hip_bf16.h>

// ---------------------------------------------------------------------------
// Shapes: B=64, C=96, H=64, W=64, Hf=64, Wf=33, N_GROUPS=4, BLOCK=24
// Workspace layout (floats):
//   xr : B*C*Hf*Wf = 12,976,128
//   xi : 12,976,128
//   y  : B*C*H*W   = 25,165,824
//   stats : 192 (sum[96], sumsq[96])
// Total ~204.5 MB.
// ---------------------------------------------------------------------------

typedef __attribute__((ext_vector_type(2))) float v2f;
typedef __attribute__((ext_vector_type(8))) float v8f;
typedef __attribute__((ext_vector_type(4))) unsigned int v4u;
typedef __attribute__((ext_vector_type(4))) int v4i;
typedef __attribute__((ext_vector_type(8))) int v8i;

__device__ __forceinline__ int brev6(int v) { return (int)(__brev((unsigned)v) >> 26); }
__device__ __forceinline__ float gelu_exact(float x) {
    return 0.5f * x * (1.0f + erff(x * 0.70710678118654752f));
}
__device__ __forceinline__ float softshrink01(float v) {
    return v > 0.01f ? (v - 0.01f) : (v < -0.01f ? (v + 0.01f) : 0.0f);
}

// ---- in-LDS 64-point radix-2 DIT passes (input must be bit-reversed) ------
// Twiddle table: twc[j]=cos(-2*pi*j/64), tws[j]=sin(-2*pi*j/64), j=0..31.
// Forward FFT: sgn = +1 (use table sin as-is). Inverse: sgn = -1 (conjugate).
__device__ __forceinline__ void fft_pass_rows(float* sre, float* sim,
                                              const float* twc, const float* tws,
                                              int tid, float sgn) {
    for (int s = 0; s < 6; ++s) {
        int mh = 1 << s;
        for (int j = tid; j < 2048; j += 256) {
            int row = j >> 5;
            int k   = j & 31;
            int pos = k & (mh - 1);
            int grp = k >> s;
            int i0 = row * 64 + grp * (mh << 1) + pos;
            int i1 = i0 + mh;
            int t  = pos << (5 - s);         // 64*pos/m
            float cs = twc[t];
            float sn = sgn * tws[t];
            float re1 = sre[i1], im1 = sim[i1];
            float tr = cs * re1 - sn * im1;
            float ti = cs * im1 + sn * re1;
            float re0 = sre[i0], im0 = sim[i0];
            sre[i1] = re0 - tr; sim[i1] = im0 - ti;
            sre[i0] = re0 + tr; sim[i0] = im0 + ti;
        }
        __syncthreads();
    }
}

__device__ __forceinline__ void fft_pass_cols(float* sre, float* sim,
                                              const float* twc, const float* tws,
                                              int tid, float sgn) {
    for (int s = 0; s < 6; ++s) {
        int mh = 1 << s;
        for (int j = tid; j < 2048; j += 256) {
            int col = j & 63;
            int k   = j >> 6;
            int pos = k & (mh - 1);
            int grp = k >> s;
            int i0 = (grp * (mh << 1) + pos) * 64 + col;
            int i1 = i0 + (mh << 6);
            int t  = pos << (5 - s);
            float cs = twc[t];
            float sn = sgn * tws[t];
            float re1 = sre[i1], im1 = sim[i1];
            float tr = cs * re1 - sn * im1;
            float ti = cs * im1 + sn * re1;
            float re0 = sre[i0], im0 = sim[i0];
            sre[i1] = re0 - tr; sim[i1] = im0 - ti;
            sre[i0] = re0 + tr; sim[i0] = im0 + ti;
        }
        __syncthreads();
    }
}

__device__ __forceinline__ void bitrev_rows(float* sre, float* sim, int tid) {
    for (int idx = tid; idx < 4096; idx += 256) {
        int h = idx >> 6, w = idx & 63;
        int hb = brev6(h);
        if (h < hb) {
            int a = h * 64 + w, b = hb * 64 + w;
            float t;
            t = sre[a]; sre[a] = sre[b]; sre[b] = t;
            t = sim[a]; sim[a] = sim[b]; sim[b] = t;
        }
    }
    __syncthreads();
}

// ---- forward rfft2 (ortho): one block per (b,c) image ---------------------
__global__ __launch_bounds__(256) void fft2_fwd(const float* __restrict__ x,
                                                float* __restrict__ xr,
                                                float* __restrict__ xi) {
    __shared__ float sre[4096];
    __shared__ float sim[4096];
    __shared__ float twc[32];
    __shared__ float tws[32];
    int img = blockIdx.x;  // b*96 + c
    int tid = threadIdx.x;
    const float* src = x + (size_t)img * 4096;
    for (int idx = tid; idx < 4096; idx += 256) {
        int h = idx >> 6, w = idx & 63;
        sre[h * 64 + brev6(w)] = src[idx];
        sim[h * 64 + brev6(w)] = 0.0f;
    }
    if (tid < 32) {
        float sn, cs;
        __sincosf(-0.09817477042468104f * (float)tid, &sn, &cs);  // -2*pi/64 * j
        twc[tid] = cs;
        tws[tid] = sn;
    }
    __syncthreads();
    fft_pass_rows(sre, sim, twc, tws, tid, 1.0f);   // FFT along W
    bitrev_rows(sre, sim, tid);
    fft_pass_cols(sre, sim, twc, tws, tid, 1.0f);   // FFT along H
    float* dr = xr + (size_t)img * 2112;
    float* di = xi + (size_t)img * 2112;
    for (int idx = tid; idx < 2112; idx += 256) {
        int h = idx / 33, w = idx - h * 33;
        dr[idx] = sre[h * 64 + w] * 0.015625f;  // 1/sqrt(64*64)
        di[idx] = sim[h * 64 + w] * 0.015625f;
    }
}

// ---- gate + softshrink + residual (low/high masks cancel pointwise) -------
__global__ __launch_bounds__(256) void gate_kernel(float* __restrict__ xr,
                                                   float* __restrict__ xi,
                                                   const float* __restrict__ w1,
                                                   const float* __restrict__ b1) {
    __shared__ float wr[576];   // w1[0][k][i][o]
    __shared__ float wi[576];   // w1[1][k][i][o]
    __shared__ float br[24], bi[24], gbr[24], gbi[24];
    int bk = blockIdx.x;
    int b = bk >> 2, k = bk & 3;
    int tid = threadIdx.x;
    for (int idx = tid; idx < 576; idx += 256) {
        wr[idx] = w1[k * 576 + idx];
        wi[idx] = w1[(4 + k) * 576 + idx];
    }
    if (tid < 24) {
        float vr = b1[k * 24 + tid];
        float vi = b1[(4 + k) * 24 + tid];
        br[tid] = vr; bi[tid] = vi;
        gbr[tid] = gelu_exact(vr);
        gbi[tid] = gelu_exact(vi);
    }
    __syncthreads();
    size_t base = ((size_t)b * 96 + (size_t)k * 24) * 2112;
    for (int p = tid; p < 2112; p += 256) {
        float vxr[24], vxi[24];
#pragma unroll
        for (int i = 0; i < 24; ++i) {
            vxr[i] = xr[base + (size_t)i * 2112 + p];
            vxi[i] = xi[base + (size_t)i * 2112 + p];
        }
#pragma unroll
        for (int o = 0; o < 24; ++o) {
            float ar = br[o], ai = bi[o];
#pragma unroll
            for (int i = 0; i < 24; ++i) {
                ar += vxr[i] * wr[i * 24 + o];
                ai += vxi[i] * wi[i * 24 + o];
            }
            float sumr = gbr[o] + gelu_exact(ar) + vxr[o];
            float sumi = gbi[o] + gelu_exact(ai) + vxi[o];
            xr[base + (size_t)o * 2112 + p] = softshrink01(sumr) + vxr[o];
            xi[base + (size_t)o * 2112 + p] = softshrink01(sumi) + vxi[o];
        }
    }
}

// ---- inverse rfft2 (ortho): Hermitian expansion + inverse FFT -------------
__global__ __launch_bounds__(256) void fft2_inv(const float* __restrict__ xr,
                                                const float* __restrict__ xi,
                                                float* __restrict__ y) {
    __shared__ float sre[4096];
    __shared__ float sim[4096];
    __shared__ float twc[32];
    __shared__ float tws[32];
    int img = blockIdx.x;
    int tid = threadIdx.x;
    const float* srcr = xr + (size_t)img * 2112;
    const float* srci = xi + (size_t)img * 2112;
    for (int idx = tid; idx < 4096; idx += 256) {
        int h = idx >> 6, w = idx & 63;
        float re, im;
        if (w < 33) {
            re = srcr[h * 33 + w];
            im = srci[h * 33 + w];
        } else {
            int h2 = (64 - h) & 63;
            int w2 = 64 - w;
            re =  srcr[h2 * 33 + w2];
            im = -srci[h2 * 33 + w2];
        }
        sre[h * 64 + brev6(w)] = re;
        sim[h * 64 + brev6(w)] = im;
    }
    if (tid < 32) {
        float sn, cs;
        __sincosf(-0.09817477042468104f * (float)tid, &sn, &cs);
        twc[tid] = cs;
        tws[tid] = sn;
    }
    __syncthreads();
    fft_pass_rows(sre, sim, twc, tws, tid, -1.0f);   // inverse: conjugate twiddles
    bitrev_rows(sre, sim, tid);
    fft_pass_cols(sre, sim, twc, tws, tid, -1.0f);
    float* dst = y + (size_t)img * 4096;
    for (int idx = tid; idx < 4096; idx += 256)
        dst[idx] = sre[idx] * 0.015625f;
}

// ---- 1x1 conv as fp32 WMMA GEMM: out[b,o,p] = sum_c W[o,c]*y[b,c,p] + pb[o]
// Block = 128 threads (4 waves). Each block: one b, one 64-wide pixel tile.
// LDS tiles are staged by the Tensor Data Mover (TENSOR_LOAD_TO_LDS), issued
// by wave 0 and synchronized with s_wait_tensorcnt + workgroup barrier.
__global__ __launch_bounds__(128) void conv1x1_wmma(const float* __restrict__ y,
                                                    const float* __restrict__ pw,
                                                    const float* __restrict__ pb,
                                                    float* __restrict__ z) {
    __shared__ float Wl[96 * 96];   // 36 KB
    __shared__ float Yl[96 * 64];   // 24 KB
    int blk = blockIdx.x;           // b*64 + ntile
    int b = blk >> 6;
    int n0 = (blk & 63) * 64;
    int tid = threadIdx.x;
    const float* ybase = y + (size_t)b * 96 * 4096 + n0;

    if (tid < 32) {
        // ---- D# for W (96x96 contiguous = one row of 9216 f32) ----
        unsigned int ldsW = (unsigned int)(size_t)(void*)&Wl[0];
        unsigned int ldsY = (unsigned int)(size_t)(void*)&Yl[0];
        unsigned long long gaW = (unsigned long long)(size_t)pw;
        unsigned long long gaY = (unsigned long long)(size_t)ybase;
        // group0: [1:0]=count=1 | lds_addr | global_addr[31:0] |
        //         global_addr[56:32] + type=2 in [127:126]
        v4u g0w = { 1u, ldsW, (unsigned int)gaW,
                    (unsigned int)((gaW >> 32) & 0x1FFFFFFull) | (2u << 30) };
        v4u g0y = { 1u, ldsY, (unsigned int)gaY,
                    (unsigned int)((gaY >> 32) & 0x1FFFFFFull) | (2u << 30) };
        // group1 (see CDNA5 ISA):
        //  w0: wg_mask=0, data_size=2 (4B)
        //  w1: barrier_addr=0 | tensor_dim0[15:0] << 16
        //  w2: tensor_dim0[31:16] | tensor_dim1[15:0] << 16
        //  w3: tensor_dim1[31:16] | tile_dim0 << 16
        //  w4: tile_dim1 | tile_dim2<<16   w5: dim0_stride[31:0]   w6,w7: 0
        v8i g1w = { (int)0x00020000,
                    (int)(9216u << 16),
                    (int)(1u << 16),
                    (int)(9216u << 16),
                    1, 9216, 0, 0 };
        v8i g1y = { (int)0x00020000,
                    (int)(4096u << 16),
                    (int)(96u << 16),
                    (int)(64u << 16),
                    96, 4096, 0, 0 };
        v4i gz = { 0, 0, 0, 0 };
#if __clang_major__ >= 23
        v8i gz8 = { 0, 0, 0, 0, 0, 0, 0, 0 };
        __builtin_amdgcn_tensor_load_to_lds(g0w, g1w, gz, gz, gz8, 0);
        __builtin_amdgcn_tensor_load_to_lds(g0y, g1y, gz, gz, gz8, 0);
#else
        __builtin_amdgcn_tensor_load_to_lds(g0w, g1w, gz, gz, 0);
        __builtin_amdgcn_tensor_load_to_lds(g0y, g1y, gz, gz, 0);
#endif
        __builtin_amdgcn_s_wait_tensorcnt(0);
    }
    // prefetch next pixel tile into cache (global_prefetch_b8)
    if (n0 + 64 < 4096) __builtin_prefetch(ybase + 64 + (size_t)(tid & 31) * 4096, 0, 0);
    __syncthreads();

    int lane = tid & 31;
    int wave = tid >> 5;
    int nw = wave * 16;                // N sub-tile offset within the 64 tile
    int m16 = lane & 15;
    int khi = (lane >> 4) << 1;        // A/B VGPR K split: lanes 16-31 hold K+2/K+3
    int mrow = (lane >> 4) << 3;       // D rows: lanes 16-31 hold M+8
    float* zbase = z + (size_t)b * 96 * 4096 + n0;

    for (int mt = 0; mt < 6; ++mt) {
        v8f acc = {};
#pragma unroll
        for (int kk = 0; kk < 24; ++kk) {
            int k0 = kk * 4;
            v2f a, bmat;
            // A (16x4): lane m holds row m; VGPR0 = K{0,2}, VGPR1 = K{1,3}
            a.x = Wl[(mt * 16 + m16) * 96 + k0 + khi];
            a.y = Wl[(mt * 16 + m16) * 96 + k0 + khi + 1];
            // B (4x16): row striped across lanes; VGPR0 = K{0,2}, VGPR1 = K{1,3}
            bmat.x = Yl[(k0 + khi) * 64 + nw + m16];
            bmat.y = Yl[(k0 + khi + 1) * 64 + nw + m16];
            acc = __builtin_amdgcn_wmma_f32_16x16x4_f32(
                false, a, false, bmat, (short)0, acc, false, false);
        }
#pragma unroll
        for (int v = 0; v < 8; ++v) {
            int m = mt * 16 + v + mrow;
            zbase[(size_t)m * 4096 + nw + m16] = acc[v] + pb[m];
        }
    }
}

// ---- BN statistics --------------------------------------------------------
__global__ void zero_stats(float* stats) {
    if (threadIdx.x < 192) stats[threadIdx.x] = 0.0f;
}

__global__ __launch_bounds__(256) void bn_stats(const float* __restrict__ z,
                                                float* __restrict__ stats) {
    int c = blockIdx.x >> 5;      // 96 channels
    int chunk = blockIdx.x & 31;  // 32 chunks of 8192 (b,p) pairs
    int tid = threadIdx.x;
    float s = 0.0f, s2 = 0.0f;
    int g0 = chunk * 8192;
    for (int g = g0 + tid; g < g0 + 8192; g += 256) {
        int bidx = g >> 12;
        int i = g & 4095;
        float v = z[((size_t)bidx * 96 + c) * 4096 + i];
        s += v;
        s2 += v * v;
    }
    __shared__ float sh[256];
    __shared__ float sh2[256];
    sh[tid] = s; sh2[tid] = s2;
    __syncthreads();
    for (int o = 128; o > 0; o >>= 1) {
        if (tid < o) { sh[tid] += sh[tid + o]; sh2[tid] += sh2[tid + o]; }
        __syncthreads();
    }
    if (tid == 0) {
        atomicAdd(&stats[c], sh[0]);
        atomicAdd(&stats[96 + c], sh2[0]);
    }
}

__global__ __launch_bounds__(256) void bn_gelu(float* __restrict__ z,
                                               const float* __restrict__ stats,
                                               const float* __restrict__ gamma,
                                               const float* __restrict__ beta) {
    size_t idx = (size_t)blockIdx.x * 256 + threadIdx.x;
    if (idx >= (size_t)25165824) return;
    int c = (int)((idx >> 12) % 96);
    const float invN = 1.0f / 262144.0f;
    float mean = stats[c] * invN;
    float var  = stats[96 + c] * invN - mean * mean;
    float inv  = rsqrtf(var + 1e-5f);
    float v = (z[idx] - mean) * inv * gamma[c] + beta[c];
    z[idx] = gelu_exact(v);
}

// ---------------------------------------------------------------------------
extern "C" void kernel_launch(void* const* d_in, const int* in_sizes, int n_in,
                              void* d_out, int out_size, void* d_ws, size_t ws_size,
                              hipStream_t stream) {
    const float* x     = (const float*)d_in[0];
    const float* w1    = (const float*)d_in[1];
    const float* b1    = (const float*)d_in[2];
    const float* pw    = (const float*)d_in[3];
    const float* pb    = (const float*)d_in[4];
    const float* gamma = (const float*)d_in[5];
    const float* beta  = (const float*)d_in[6];
    float* out = (float*)d_out;
    float* ws  = (float*)d_ws;

    float* xr    = ws;
    float* xi    = ws + 12976128;
    float* ybuf  = ws + 2 * 12976128;
    float* stats = ybuf + 25165824;

    fft2_fwd<<<6144, 256, 0, stream>>>(x, xr, xi);
    gate_kernel<<<256, 256, 0, stream>>>(xr, xi, w1, b1);
    fft2_inv<<<6144, 256, 0, stream>>>(xr, xi, ybuf);
    conv1x1_wmma<<<4096, 128, 0, stream>>>(ybuf, pw, pb, out);
    zero_stats<<<1, 256, 0, stream>>>(stats);
    bn_stats<<<96 * 32, 256, 0, stream>>>(out, stats);
    bn_gelu<<<98304, 256, 0, stream>>>(out, stats, gamma, beta);
}